// myConv_21010980012292
// MI455X (gfx1250) — compile-verified
//
#include <hip/hip_runtime.h>

// 7x7 stride-2 VALID conv on 4096x4096 f32 -> 2045x2045 f32 via
// V_WMMA_F32_16X16X4_F32. Memory-bound (~80 MB @ 23.3 TB/s ~ 3.5 us).
// Kernel rows padded to 8 taps (zero weight at kx=7) so every LDS fragment
// address is a compile-time immediate: one ds_load_b64 per K=4 chunk.

#define HDIM    4096
#define OUTDIM  2045
#define OTY     16
#define OTX     16
#define ITY     37          // 2*OTY + 7 - 2
#define LDSW    40          // padded LDS row stride in floats (10 x float4)
#define NVEC    (ITY * 10)  // float4 loads per tile

typedef __attribute__((ext_vector_type(2))) float v2f;
typedef __attribute__((ext_vector_type(8))) float v8f;
typedef __attribute__((ext_vector_type(4))) float f4;

__global__ __launch_bounds__(256) void conv7x7s2_wmma_kernel(
    const float* __restrict__ X, const float* __restrict__ W,
    float* __restrict__ out)
{
    __shared__ float Xs[ITY * LDSW];
    __shared__ float Ws[56];            // 7 rows x 8 taps, kx=7 is zero

    const int tid = threadIdx.x;
    const int ox0 = blockIdx.x * OTX;
    const int oy0 = blockIdx.y * OTY;
    const int ix0 = ox0 * 2;            // multiple of 32 -> 128B aligned
    const int iy0 = oy0 * 2;

    // Stage weights, row-padded to 8 (zero in the pad column).
    if (tid < 56) {
        const int ky = tid >> 3, kx = tid & 7;
        Ws[tid] = (kx < 7) ? W[ky * 7 + kx] : 0.0f;
    }

    // Stage 37-row x 40-col input tile into LDS with 128-bit loads.
    for (int i = tid; i < NVEC; i += 256) {
        const int r  = i / 10;
        const int c4 = (i % 10) * 4;
        const int gy = iy0 + r;
        const int gx = ix0 + c4;
        f4 v;
        if (gy < HDIM && (gx + 3) < HDIM) {
            v = *reinterpret_cast<const f4*>(X + (size_t)gy * HDIM + gx);
        } else {
            v.x = (gy < HDIM && (gx + 0) < HDIM) ? X[(size_t)gy * HDIM + gx + 0] : 0.0f;
            v.y = (gy < HDIM && (gx + 1) < HDIM) ? X[(size_t)gy * HDIM + gx + 1] : 0.0f;
            v.z = (gy < HDIM && (gx + 2) < HDIM) ? X[(size_t)gy * HDIM + gx + 2] : 0.0f;
            v.w = (gy < HDIM && (gx + 3) < HDIM) ? X[(size_t)gy * HDIM + gx + 3] : 0.0f;
        }
        *reinterpret_cast<f4*>(&Xs[r * LDSW + c4]) = v;
    }
    __syncthreads();

    const int lane = tid & 31;
    const int wave = tid >> 5;
    const int h    = lane >> 4;   // wave half: VGPR j holds K = 2h + j
    const int n    = lane & 15;   // output column within the 16-wide strip

    // A fragments (weights, shared by both output rows): Ws[8ky + 4c + 2h + j]
    const float* ap = &Ws[2 * h];
    v2f a[14];
    #pragma unroll
    for (int ky = 0; ky < 7; ++ky) {
        #pragma unroll
        for (int c = 0; c < 2; ++c) {
            a[ky * 2 + c] = *reinterpret_cast<const v2f*>(ap + 8 * ky + 4 * c);
        }
    }

    // B base for this lane: Xs[(4*wave)*LDSW + 2n + 2h]; all tap offsets are
    // compile-time constants -> ds_load_b64 with immediate offsets.
    const float* bp = &Xs[(4 * wave) * LDSW + 2 * n + 2 * h];

    #pragma unroll
    for (int rr = 0; rr < 2; ++rr) {
        v8f acc = {};
        #pragma unroll
        for (int ky = 0; ky < 7; ++ky) {
            #pragma unroll
            for (int c = 0; c < 2; ++c) {
                const v2f b = *reinterpret_cast<const v2f*>(
                    bp + rr * 2 * LDSW + ky * LDSW + 4 * c);
                acc = __builtin_amdgcn_wmma_f32_16x16x4_f32(
                    false, a[ky * 2 + c], false, b, (short)0, acc, false, false);
            }
        }
        if (h == 0) {
            const int oy = oy0 + wave * 2 + rr;
            const int ox = ox0 + n;
            if (oy < OUTDIM && ox < OUTDIM)
                out[(size_t)oy * OUTDIM + ox] = acc[0];
        }
    }
}

extern "C" void kernel_launch(void* const* d_in, const int* in_sizes, int n_in,
                              void* d_out, int out_size, void* d_ws, size_t ws_size,
                              hipStream_t stream)
{
    (void)in_sizes; (void)n_in; (void)d_ws; (void)ws_size; (void)out_size;
    const float* X = (const float*)d_in[0];
    const float* W = (const float*)d_in[1];
    float* out = (float*)d_out;

    dim3 grid((OUTDIM + OTX - 1) / OTX, (OUTDIM + OTY - 1) / OTY);
    dim3 block(256);
    conv7x7s2_wmma_kernel<<<grid, block, 0, stream>>>(X, W, out);
}